// HierarchicalModel_86835648790828
// MI455X (gfx1250) — compile-verified
//
#include <hip/hip_runtime.h>
#include <math.h>

// Problem constants (from setup_inputs)
#define Bsz 16
#define Tsz 1024
#define Dsz 96
#define Ksz 16
#define Psz 64
#define TT  64            // timesteps per workgroup tile in the WMMA kernel

typedef float v2f __attribute__((ext_vector_type(2)));
typedef float v8f __attribute__((ext_vector_type(8)));

// Workspace layout (in floats)
#define WS_NLL    0        // 256 banked NLL partial accumulators
#define WS_MUREG  256      // 1
#define WS_LREG   257      // 1
#define WS_MASK   258      // 64  present-subject mask
#define WS_LOGDET 322      // 256 logdet per (b,k)
#define WS_LINV   640      // 256 * 96*96 floats = 9.44 MB of Linv scratch

// 0.5*log(2*pi) * D
#define D_HALF_LOG2PI 88.2180991876486f

// ---------------------------------------------------------------------------
// Kernel 0: zero accumulators, compute present-subject mask
// ---------------------------------------------------------------------------
__global__ void init_kernel(float* __restrict__ ws, const int* __restrict__ sid) {
    int t = threadIdx.x;
    for (int i = t; i < 258; i += 256) ws[i] = 0.0f;
    if (t < Psz) {
        float m = 0.0f;
        for (int b = 0; b < Bsz; ++b)
            if (sid[b] == t) m = 1.0f;
        ws[WS_MASK + t] = m;
    }
}

// ---------------------------------------------------------------------------
// Kernel 1: per (b,k) lower-triangular inverse (forward substitution, thread
// owns one column of the identity RHS -> no cross-thread deps) + logdet.
// grid = (K, B), block = 96 threads.
// ---------------------------------------------------------------------------
__global__ void invert_kernel(const float* __restrict__ L_subj,
                              const int* __restrict__ sid,
                              float* __restrict__ ws) {
    __shared__ float sXc[Dsz][Dsz + 1];   // Linv columns; [row][col], col = tid
    __shared__ float sRed[Dsz];
    const int k = blockIdx.x;
    const int b = blockIdx.y;
    const int p = sid[b];
    const float* Lg = L_subj + ((size_t)p * Ksz + k) * (Dsz * Dsz);
    const int j = threadIdx.x;            // column 0..95

    // Solve L * X = I, column j.  Each thread touches only column j of sXc.
    for (int d = 0; d < Dsz; ++d) {
        float acc = (j == d) ? 1.0f : 0.0f;
        for (int i = 0; i < d; ++i)
            acc -= Lg[d * Dsz + i] * sXc[i][j];   // Lg read is uniform (scalar)
        sXc[d][j] = acc / Lg[d * Dsz + d];
    }
    sRed[j] = logf(Lg[j * Dsz + j]);
    __syncthreads();

    if (j == 0) {
        float s = 0.0f;
        for (int i = 0; i < Dsz; ++i) s += sRed[i];
        ws[WS_LOGDET + b * Ksz + k] = s;
    }
    // Write Linv row-major to workspace
    float* dst = ws + WS_LINV + (size_t)(b * Ksz + k) * (Dsz * Dsz);
    for (int i = j; i < Dsz * Dsz; i += Dsz)
        dst[i] = sXc[i / Dsz][i % Dsz];
}

// ---------------------------------------------------------------------------
// Kernel 2: the hot kernel.  One workgroup = (b, k, 64-wide t-tile).
// Y = Linv @ R via V_WMMA_F32_16X16X4_F32, quad = sum_d Y^2 accumulated
// in-register from the D fragments, then gamma-weighted NLL accumulation.
// block = 256 threads (8 waves).
// Wave w: t-subblock (w&3), M-half (w>>2) -> 3 M-blocks of 16 rows each.
// ---------------------------------------------------------------------------
__launch_bounds__(256, 1)
__global__ void nll_kernel(const float* __restrict__ x,
                           const float* __restrict__ gamma,
                           const int* __restrict__ sid,
                           const float* __restrict__ mu_subj,
                           float* __restrict__ ws) {
    __shared__ float sL[Dsz][Dsz + 2];   // Linv tile, row stride 98 (conflict-free)
    __shared__ float sR[TT][Dsz + 2];    // residuals (t-local, d), row stride 98
    __shared__ float sQ[8][16];          // per-wave quad partials

    const int t0  = blockIdx.x * TT;
    const int k   = blockIdx.y;
    const int b   = blockIdx.z;
    const int tid = threadIdx.x;
    const int p   = sid[b];

    const float* Lbase = ws + WS_LINV + (size_t)(b * Ksz + k) * (Dsz * Dsz);
    const float* mu    = mu_subj + ((size_t)p * Ksz + k) * Dsz;

    for (int i = tid; i < Dsz * Dsz; i += 256)
        sL[i / Dsz][i % Dsz] = Lbase[i];
    for (int i = tid; i < TT * Dsz; i += 256) {
        int tt = i / Dsz, d = i - tt * Dsz;
        sR[tt][d] = x[((size_t)b * Tsz + (t0 + tt)) * Dsz + d] - mu[d];
    }
    __syncthreads();

    const int wave  = tid >> 5;
    const int lane  = tid & 31;
    const int l16   = lane & 15;
    const int h     = lane >> 4;     // half-wave select
    const int nblk  = wave & 3;      // which 16-wide t sub-block
    const int mhalf = wave >> 2;     // rows 0..47 vs 48..95

    // A fragment (16x4 f32): lane row = l16, per-lane K = {2h, 2h+1}
    // B fragment (4x16 f32): lane col = l16, per-lane K = {2h, 2h+1}
    // D fragment (16x16 f32): lane col = l16, rows = v + 8h, v = 0..7
    const float* brow = &sR[nblk * 16 + l16][0];
    float qacc = 0.0f;

    for (int mb = mhalf * 3; mb < mhalf * 3 + 3; ++mb) {
        v8f acc = {0.f, 0.f, 0.f, 0.f, 0.f, 0.f, 0.f, 0.f};
        const float* arow = &sL[mb * 16 + l16][0];
#pragma unroll
        for (int k0 = 0; k0 < Dsz; k0 += 4) {
            v2f a, bb;
            a[0]  = arow[k0 + 2 * h];
            a[1]  = arow[k0 + 2 * h + 1];
            bb[0] = brow[k0 + 2 * h];
            bb[1] = brow[k0 + 2 * h + 1];
            acc = __builtin_amdgcn_wmma_f32_16x16x4_f32(
                /*neg_a=*/false, a, /*neg_b=*/false, bb,
                /*c_mod=*/(short)0, acc, /*reuse_a=*/false, /*reuse_b=*/false);
        }
#pragma unroll
        for (int v = 0; v < 8; ++v) qacc += acc[v] * acc[v];
    }

    // combine the two half-wave row groups, then the two M-half waves
    qacc += __shfl_xor(qacc, 16, 32);
    if (lane < 16) sQ[wave][lane] = qacc;
    __syncthreads();

    float contrib = 0.0f;
    if (wave < 4 && lane < 16) {
        float q = sQ[wave][lane] + sQ[wave + 4][lane];
        int t = t0 + wave * 16 + lane;
        float g = gamma[((size_t)b * Tsz + t) * Ksz + k];
        float logdet = ws[WS_LOGDET + b * Ksz + k];
        contrib = g * (0.5f * q + logdet + D_HALF_LOG2PI);
    }
#pragma unroll
    for (int off = 16; off >= 1; off >>= 1)
        contrib += __shfl_xor(contrib, off, 32);
    if (wave < 4 && lane == 0) {
        int slot = (blockIdx.x + blockIdx.y * 16 + blockIdx.z) & 255;
        atomicAdd(&ws[WS_NLL + slot], contrib);
    }
}

// ---------------------------------------------------------------------------
// Kernel 3: masked Frobenius regularizer sums (grid-stride).
// ---------------------------------------------------------------------------
__global__ void reg_kernel(const float* __restrict__ mu_pop,
                           const float* __restrict__ L_pop,
                           const float* __restrict__ mu_subj,
                           const float* __restrict__ L_subj,
                           float* __restrict__ ws) {
    const int N1 = Psz * Ksz * Dsz;          // 98,304
    const int N2 = Psz * Ksz * Dsz * Dsz;    // 9,437,184
    const int stride = gridDim.x * blockDim.x;
    int gid = blockIdx.x * blockDim.x + threadIdx.x;
    float v = 0.0f, w = 0.0f;
    for (int i = gid; i < N1; i += stride) {
        int p = i / (Ksz * Dsz);
        int rem = i - p * (Ksz * Dsz);
        float d = mu_subj[i] - mu_pop[rem];
        v += ws[WS_MASK + p] * d * d;
    }
    for (int i = gid; i < N2; i += stride) {
        int p = i / (Ksz * Dsz * Dsz);
        int rem = i - p * (Ksz * Dsz * Dsz);
        float d = L_subj[i] - L_pop[rem];
        w += ws[WS_MASK + p] * d * d;
    }
    __shared__ float s1[256], s2[256];
    s1[threadIdx.x] = v; s2[threadIdx.x] = w;
    __syncthreads();
    for (int o = 128; o >= 1; o >>= 1) {
        if (threadIdx.x < o) {
            s1[threadIdx.x] += s1[threadIdx.x + o];
            s2[threadIdx.x] += s2[threadIdx.x + o];
        }
        __syncthreads();
    }
    if (threadIdx.x == 0) {
        atomicAdd(&ws[WS_MUREG], s1[0]);
        atomicAdd(&ws[WS_LREG],  s2[0]);
    }
}

// ---------------------------------------------------------------------------
// Kernel 4: final combine into scalar output.
// ---------------------------------------------------------------------------
__global__ void finalize_kernel(const float* __restrict__ ws,
                                float* __restrict__ out) {
    __shared__ float s[256];
    __shared__ float sm[64];
    int t = threadIdx.x;
    s[t] = ws[WS_NLL + t];
    if (t < 64) sm[t] = ws[WS_MASK + t];
    __syncthreads();
    for (int o = 128; o >= 1; o >>= 1) {
        if (t < o) s[t] += s[t + o];
        __syncthreads();
    }
    if (t == 0) {
        float S = 0.0f;
        for (int i = 0; i < Psz; ++i) S += sm[i];
        float nll = s[0] / (float)(Bsz * Tsz);
        float reg = (S / (float)Psz) *
                    (0.05f * ws[WS_MUREG] + 0.05f * ws[WS_LREG]); // LAMBDA/2 = 0.05
        out[0] = nll + reg;
    }
}

// ---------------------------------------------------------------------------
extern "C" void kernel_launch(void* const* d_in, const int* in_sizes, int n_in,
                              void* d_out, int out_size, void* d_ws, size_t ws_size,
                              hipStream_t stream) {
    const float* x       = (const float*)d_in[0];
    const float* gamma   = (const float*)d_in[1];
    const int*   sid     = (const int*)  d_in[2];
    const float* mu_pop  = (const float*)d_in[3];
    const float* L_pop   = (const float*)d_in[4];
    const float* mu_subj = (const float*)d_in[5];
    const float* L_subj  = (const float*)d_in[6];
    float* ws  = (float*)d_ws;
    float* out = (float*)d_out;

    init_kernel<<<1, 256, 0, stream>>>(ws, sid);

    dim3 gInv(Ksz, Bsz);
    invert_kernel<<<gInv, Dsz, 0, stream>>>(L_subj, sid, ws);

    dim3 gMain(Tsz / TT, Ksz, Bsz);   // (16, 16, 16) = 4096 workgroups
    nll_kernel<<<gMain, 256, 0, stream>>>(x, gamma, sid, mu_subj, ws);

    reg_kernel<<<1024, 256, 0, stream>>>(mu_pop, L_pop, mu_subj, L_subj, ws);

    finalize_kernel<<<1, 256, 0, stream>>>(ws, out);
}